// StochasticLoss_59304908423657
// MI455X (gfx1250) — compile-verified
//
#include <hip/hip_runtime.h>
#include <stdint.h>

// ---------------------------------------------------------------------------
// StochasticLoss: p = mean_s sigmoid(logit + std*noise[s]); BCE(mean) -> scalar
// N = 1e6 rows, S = 100 samples. HBM-bound: 412 MB / 23.3 TB/s ~ 17.7 us floor.
// Strategy: one wave per row (lanes 0..24 hold one float4 each). Noise rows are
// DMA'd global->LDS by the CDNA5 Tensor Data Mover (tensor_load_to_lds, one
// instruction per G-row group per wave, TENSORcnt double-buffered). Sigmoid via
// hardware V_TANH_F32: sigmoid(x) = 0.5 + 0.5*tanh(x/2), 1 trans op/sample.
// Fallback: per-lane global_load_async_to_lds_b128 (ASYNCcnt) if no builtin.
// ---------------------------------------------------------------------------

#define S_SAMPLES 100
#define ROW_BYTES 400               // 100 f32
#define BLOCK 256
#define WAVES_PER_BLOCK 8           // wave32
#define G 8                         // rows per pipeline group per wave
#define GRID 4096
#define WAVE_STRIDE (GRID * WAVES_PER_BLOCK)   // 32768 (power of two)

#define STR2(x) #x
#define STR(x) STR2(x)

typedef unsigned int v4u __attribute__((ext_vector_type(4)));
typedef int          v8i __attribute__((ext_vector_type(8)));
typedef int          v4i __attribute__((ext_vector_type(4)));

#if __has_builtin(__builtin_amdgcn_tensor_load_to_lds)
#define USE_TDM 1
#else
#define USE_TDM 0
#endif

__device__ __forceinline__ float fast_sigmoid(float x) {
    // sigmoid(x) = 0.5 * (1 + tanh(x/2)) -> single V_TANH_F32 trans op
#if __has_builtin(__builtin_amdgcn_tanhf)
    return fmaf(0.5f, __builtin_amdgcn_tanhf(0.5f * x), 0.5f);
#else
    float t;
    // v_nop after trans op: ISA 7.4 multicycle co-execution hazard rule
    asm("v_tanh_f32 %0, %1\n\tv_nop" : "=v"(t) : "v"(0.5f * x));
    return fmaf(0.5f, t, 0.5f);
#endif
}

// Issue one group: `rows` noise rows (row indices rowStart + j*WAVE_STRIDE)
// into LDS at byte offset ldsDst (rows packed contiguously, 400 B apart).
__device__ __forceinline__ void issue_group(const float* __restrict__ noise,
                                            unsigned rowStart, unsigned rows,
                                            unsigned ldsDst, unsigned lane)
{
#if USE_TDM
    // ---- Tensor DMA descriptor (ISA 8.3/8.4), 2-D tile, data_size = 4 B ----
    uint64_t gaddr = (uint64_t)(uintptr_t)noise + (uint64_t)rowStart * ROW_BYTES;
    v4u g0;
    g0.x = 1u;                                            // count=1 (valid D#)
    g0.y = ldsDst;                                        // lds_addr (bytes)
    g0.z = (unsigned)gaddr;                               // global_addr[31:0]
    g0.w = (unsigned)((gaddr >> 32) & 0x1FFFFFFu)         // global_addr[56:32]
         | (2u << 30);                                    // type=2 ("image")
    v8i g1;
    g1[0] = (int)(2u << 16);                              // data_size=2 -> 4B
    g1[1] = (int)((unsigned)S_SAMPLES << 16);             // tensor_dim0[15:0]=100
    g1[2] = (int)(rows << 16);                            // td0 hi=0 | tensor_dim1[15:0]
    g1[3] = (int)((unsigned)S_SAMPLES << 16);             // td1 hi=0 | tile_dim0=100
    g1[4] = (int)rows;                                    // tile_dim1=rows, tile_dim2=0
    g1[5] = (int)((unsigned)WAVE_STRIDE * S_SAMPLES);     // tensor_dim0_stride[31:0]
    g1[6] = 0;                                            // stride0 hi | stride1 lo
    g1[7] = 0;                                            // stride1 hi
    v4i gz = {0, 0, 0, 0};
    (void)lane;
#if defined(__clang_major__) && (__clang_major__ >= 23)
    v8i gz8 = {0, 0, 0, 0, 0, 0, 0, 0};
    __builtin_amdgcn_tensor_load_to_lds(g0, g1, gz, gz, gz8, 0);
#else
    __builtin_amdgcn_tensor_load_to_lds(g0, g1, gz, gz, 0);
#endif
#else
    // ---- fallback: per-lane async global->LDS (ASYNCcnt) -------------------
    for (unsigned j = 0; j < rows; ++j) {
        unsigned r = rowStart + j * (unsigned)WAVE_STRIDE;
        if (lane < 25u) {
            unsigned voff = r * ROW_BYTES + lane * 16u;
            unsigned dst  = ldsDst + j * ROW_BYTES + lane * 16u;
            asm volatile("global_load_async_to_lds_b128 %0, %1, %2"
                         :: "v"(dst), "v"(voff), "s"(noise) : "memory");
        }
    }
#endif
}

// Wait until the *current* group's DMA has landed in LDS.
// pendingNext==true -> one more group (1 TDM op / G async ops) may stay in flight.
__device__ __forceinline__ void wait_group(bool pendingNext)
{
#if USE_TDM
    if (pendingNext) __builtin_amdgcn_s_wait_tensorcnt((short)1);
    else             __builtin_amdgcn_s_wait_tensorcnt((short)0);
#else
    if (pendingNext) asm volatile("s_wait_asynccnt " STR(G) ::: "memory");
    else             asm volatile("s_wait_asynccnt 0" ::: "memory");
#endif
    asm volatile("" ::: "memory");   // compiler barrier: no LDS-read hoisting
}

__global__ __launch_bounds__(BLOCK) void StochasticLoss_main_kernel(
    const float* __restrict__ logits,
    const float* __restrict__ stddevs,
    const float* __restrict__ targets,
    const float* __restrict__ noise,
    double* __restrict__ acc_ws,
    unsigned int N)
{
    __shared__ float lds[WAVES_PER_BLOCK * 2 * G * S_SAMPLES];   // 51.2 KB
    __shared__ float waveAcc[WAVES_PER_BLOCK];

    const unsigned lane = threadIdx.x & 31u;
    const unsigned wave = threadIdx.x >> 5;
    const unsigned waveGlobal = blockIdx.x * WAVES_PER_BLOCK + wave;

    float* myLds = &lds[wave * 2 * G * S_SAMPLES];
    // gfx1250: low 32 bits of a flat LDS pointer == LDS byte offset (ISA 10.2)
    const unsigned ldsBase = (unsigned)(uintptr_t)myLds;

    // WAVE_STRIDE is a power-of-two compile-time constant -> shift, not divide
    unsigned myRows = 0;
    if (waveGlobal < N) myRows = (N - 1u - waveGlobal) / (unsigned)WAVE_STRIDE + 1u;

    float acc = 0.0f;

    unsigned k = 0, half = 0;
    {
        unsigned first = myRows < (unsigned)G ? myRows : (unsigned)G;
        if (first) issue_group(noise, waveGlobal, first, ldsBase, lane);
    }

    while (k < myRows) {
        unsigned cur = myRows - k;  if (cur > (unsigned)G) cur = (unsigned)G;
        unsigned nk  = k + cur;
        unsigned nxt = 0;
        if (nk < myRows) { nxt = myRows - nk; if (nxt > (unsigned)G) nxt = (unsigned)G; }

        // prefetch next group into the other LDS half while we consume this one
        if (nxt) {
            issue_group(noise, waveGlobal + nk * (unsigned)WAVE_STRIDE, nxt,
                        ldsBase + (half ^ 1u) * G * ROW_BYTES, lane);
            wait_group(true);      // current group done; next may stay in flight
        } else {
            wait_group(false);     // drain
        }

        for (unsigned j = 0; j < cur; ++j) {
            unsigned r = waveGlobal + (k + j) * (unsigned)WAVE_STRIDE;
            float l  = logits[r];      // broadcast loads (uniform address)
            float sd = stddevs[r];
            float part = 0.0f;
            if (lane < 25u) {
                const float4 nv = *reinterpret_cast<const float4*>(
                    &myLds[(half * G + j) * S_SAMPLES + lane * 4u]);
                part  = fast_sigmoid(fmaf(sd, nv.x, l));
                part += fast_sigmoid(fmaf(sd, nv.y, l));
                part += fast_sigmoid(fmaf(sd, nv.z, l));
                part += fast_sigmoid(fmaf(sd, nv.w, l));
            }
            // wave32 butterfly reduction
            part += __shfl_xor(part, 16, 32);
            part += __shfl_xor(part,  8, 32);
            part += __shfl_xor(part,  4, 32);
            part += __shfl_xor(part,  2, 32);
            part += __shfl_xor(part,  1, 32);
            if (lane == 0u) {
                float p      = part * (1.0f / (float)S_SAMPLES);
                float t      = targets[r];
                float logp   = fmaxf(__logf(p),  -100.0f);
                float log1mp = fmaxf(log1pf(-p), -100.0f);
                acc -= t * logp + (1.0f - t) * log1mp;
            }
        }
        k = nk;
        half ^= 1u;
    }

    if (lane == 0u) waveAcc[wave] = acc;
    __syncthreads();
    if (threadIdx.x == 0u) {
        double s = 0.0;
        #pragma unroll
        for (int i = 0; i < WAVES_PER_BLOCK; ++i) s += (double)waveAcc[i];
        atomicAdd(acc_ws, s);   // global_atomic_add_f64
    }
}

__global__ void StochasticLoss_finalize_kernel(const double* __restrict__ acc,
                                               float* __restrict__ out,
                                               double invN)
{
    if (threadIdx.x == 0u && blockIdx.x == 0u) {
        out[0] = (float)(acc[0] * invN);
    }
}

extern "C" void kernel_launch(void* const* d_in, const int* in_sizes, int n_in,
                              void* d_out, int out_size, void* d_ws, size_t ws_size,
                              hipStream_t stream)
{
    const float* logits  = (const float*)d_in[0];
    const float* stddevs = (const float*)d_in[1];
    const float* targets = (const float*)d_in[2];
    const float* noise   = (const float*)d_in[3];
    const unsigned N = (unsigned)in_sizes[0];

    double* acc = (double*)d_ws;
    hipMemsetAsync(d_ws, 0, sizeof(double), stream);   // graph-capture safe

    StochasticLoss_main_kernel<<<GRID, BLOCK, 0, stream>>>(
        logits, stddevs, targets, noise, acc, N);

    StochasticLoss_finalize_kernel<<<1, 32, 0, stream>>>(
        acc, (float*)d_out, 1.0 / (double)N);
}